// GATBackbone_33131377721477
// MI455X (gfx1250) — compile-verified
//
#include <hip/hip_runtime.h>
#include <math.h>

#define NNODES   50000
#define NEDGES   1600000
#define HC       128          // HEADS * HID
#define NHEADS   4
#define NHID     32
#define ETOT     (NEDGES + NNODES)
#define NEGSLOPE 0.2f

typedef float v2f __attribute__((ext_vector_type(2)));
typedef float v8f __attribute__((ext_vector_type(8)));

// ---------------------------------------------------------------------------
// Y[M x 128] = X[M x 128] @ W[128 x 128] + bias   (f32 WMMA 16x16x4)
// 256 threads = 8 wave32. Each wave computes one 16-row strip x 128 cols.
// A frag (16x4 f32, 2 VGPR): lanes 0-15 hold K={k0,k0+1}, lanes 16-31 K={k0+2,k0+3}.
// B frag (4x16 f32, 2 VGPR): same K split, N = lane%16.
// C/D (16x16 f32, 8 VGPR): VGPR i -> (M=i or i+8, N=lane%16).
// ---------------------------------------------------------------------------
__global__ __launch_bounds__(256) void gemm128_wmma(
    const float* __restrict__ X, const float* __restrict__ W,
    const float* __restrict__ bias, float* __restrict__ Y, int nWaveTiles)
{
  __shared__ float sW[128 * 128];                      // 64 KB
  const int tid = threadIdx.x;
  for (int i = tid; i < 128 * 128; i += 256) sW[i] = W[i];
  __syncthreads();

  const int waveIdx = blockIdx.x * 8 + (tid >> 5);
  if (waveIdx >= nWaveTiles) return;                   // whole-wave exit: EXEC stays all-1s
  const int lane  = tid & 31;
  const int mrow  = lane & 15;
  const int khalf = lane >> 4;                         // 0 or 1
  const int row0  = waveIdx * 16;
  const float* __restrict__ xrow = X + (size_t)(row0 + mrow) * HC;

  v8f acc[8] = {};
  for (int k0 = 0; k0 < 128; k0 += 4) {
    const int ka = k0 + khalf * 2;
    v2f a; a.x = xrow[ka]; a.y = xrow[ka + 1];
#pragma unroll
    for (int j = 0; j < 8; ++j) {
      v2f b;
      b.x = sW[(ka + 0) * 128 + j * 16 + mrow];
      b.y = sW[(ka + 1) * 128 + j * 16 + mrow];
      acc[j] = __builtin_amdgcn_wmma_f32_16x16x4_f32(
          false, a, false, b, (short)0, acc[j], false, false);
    }
  }
#pragma unroll
  for (int j = 0; j < 8; ++j) {
    const int col = j * 16 + mrow;
    const float bj = bias[col];
#pragma unroll
    for (int i = 0; i < 8; ++i) {
      const int r = row0 + i + 8 * khalf;
      Y[(size_t)r * HC + col] = acc[j][i] + bj;
    }
  }
}

// ---------------------------------------------------------------------------
// Per-layer state init: m = -inf, z = 0, acc = 0
// ---------------------------------------------------------------------------
__global__ __launch_bounds__(256) void init_layer(
    float* __restrict__ m, float* __restrict__ z, float* __restrict__ acc)
{
  const int t = blockIdx.x * 256 + threadIdx.x;
  if (t < NNODES * NHEADS) { m[t] = -INFINITY; z[t] = 0.0f; }
  if (t < NNODES * HC) acc[t] = 0.0f;
}

// ---------------------------------------------------------------------------
// One wave per edge. Lane l covers channel l of each head (HID == 32 == wave32).
// s[e][h] = sum_k att[h][k] * leakyrelu(xl[src][h][k] + xr[dst][h][k])
// then atomicMax into m[dst][h].
// ---------------------------------------------------------------------------
__global__ __launch_bounds__(256) void edge_scores(
    const long long* __restrict__ ei, const float* __restrict__ xl,
    const float* __restrict__ xr, const float* __restrict__ att,
    float* __restrict__ s, float* __restrict__ m)
{
  const int e = blockIdx.x * 8 + (threadIdx.x >> 5);
  if (e >= ETOT) return;
  const int lane = threadIdx.x & 31;
  int src, dst;
  if (e < NEDGES) { src = (int)ei[e]; dst = (int)ei[NEDGES + e]; }
  else            { src = e - NEDGES; dst = src; }

  const float* __restrict__ xls = xl + (size_t)src * HC;
  const float* __restrict__ xrd = xr + (size_t)dst * HC;
  float sv[NHEADS];
#pragma unroll
  for (int j = 0; j < NHEADS; ++j) {
    const float h  = xls[j * NHID + lane] + xrd[j * NHID + lane];
    const float lr = (h > 0.0f) ? h : NEGSLOPE * h;
    float v = lr * att[j * NHID + lane];
#pragma unroll
    for (int off = 16; off > 0; off >>= 1) v += __shfl_xor(v, off, 32);
    sv[j] = v;
  }
  if (lane == 0) {
#pragma unroll
    for (int j = 0; j < NHEADS; ++j) {
      s[(size_t)e * NHEADS + j] = sv[j];
      atomicMax(&m[(size_t)dst * NHEADS + j], sv[j]);
    }
  }
}

// ---------------------------------------------------------------------------
// Thread per (edge, head): w = exp(s - m[dst]) (in place), atomicAdd into z[dst].
// ---------------------------------------------------------------------------
__global__ __launch_bounds__(256) void edge_expsum(
    const long long* __restrict__ ei, float* __restrict__ s,
    const float* __restrict__ m, float* __restrict__ z)
{
  const int t = blockIdx.x * 256 + threadIdx.x;
  if (t >= ETOT * NHEADS) return;
  const int e = t >> 2, j = t & 3;
  const int dst = (e < NEDGES) ? (int)ei[NEDGES + e] : (e - NEDGES);
  const float w = expf(s[t] - m[(size_t)dst * NHEADS + j]);
  s[t] = w;
  atomicAdd(&z[(size_t)dst * NHEADS + j], w);
}

// ---------------------------------------------------------------------------
// One wave per edge: acc[dst][h][k] += (w[e][h]/z[dst][h]) * xl[src][h][k]
// ---------------------------------------------------------------------------
__global__ __launch_bounds__(256) void edge_aggregate(
    const long long* __restrict__ ei, const float* __restrict__ xl,
    const float* __restrict__ w, const float* __restrict__ z,
    float* __restrict__ acc)
{
  const int e = blockIdx.x * 8 + (threadIdx.x >> 5);
  if (e >= ETOT) return;
  const int lane = threadIdx.x & 31;
  int src, dst;
  if (e < NEDGES) { src = (int)ei[e]; dst = (int)ei[NEDGES + e]; }
  else            { src = e - NEDGES; dst = src; }

  const float* __restrict__ xls = xl + (size_t)src * HC;
  float* __restrict__ accd = acc + (size_t)dst * HC;
#pragma unroll
  for (int j = 0; j < NHEADS; ++j) {
    const float alpha = w[(size_t)e * NHEADS + j] / z[(size_t)dst * NHEADS + j];
    atomicAdd(&accd[j * NHID + lane], alpha * xls[j * NHID + lane]);
  }
}

// ---------------------------------------------------------------------------
// out = acc + b, optional ELU
// ---------------------------------------------------------------------------
__global__ __launch_bounds__(256) void finalize(
    const float* __restrict__ acc, const float* __restrict__ b,
    float* __restrict__ out, int applyElu)
{
  const int t = blockIdx.x * 256 + threadIdx.x;
  if (t >= NNODES * HC) return;
  float v = acc[t] + b[t & (HC - 1)];
  if (applyElu) v = (v > 0.0f) ? v : (expf(v) - 1.0f);
  out[t] = v;
}

// ---------------------------------------------------------------------------
extern "C" void kernel_launch(void* const* d_in, const int* in_sizes, int n_in,
                              void* d_out, int out_size, void* d_ws, size_t ws_size,
                              hipStream_t stream)
{
  (void)in_sizes; (void)n_in; (void)out_size; (void)ws_size;
  const float*      x0 = (const float*)d_in[0];
  const long long*  ei = (const long long*)d_in[1];   // int64 edge_index

  float* ws = (float*)d_ws;
  const size_t NF = (size_t)NNODES * HC;              // 6.4M floats
  const size_t EH = (size_t)ETOT * NHEADS;            // 6.6M floats
  float* xl  = ws;
  float* xr  = xl  + NF;
  float* acc = xr  + NF;
  float* xnx = acc + NF;
  float* sw  = xnx + NF;                              // scores, then weights (in place)
  float* m   = sw  + EH;
  float* z   = m   + (size_t)NNODES * NHEADS;

  const int  nWaveTiles = NNODES / 16;                // 3125 exactly
  const dim3 gemmGrid((nWaveTiles + 7) / 8);
  const dim3 edgeGrid((ETOT + 7) / 8);                // 8 wave32 edges per block
  const dim3 nodeGrid((unsigned)((NF + 255) / 256));
  const dim3 ehGrid((unsigned)(((size_t)ETOT * NHEADS + 255) / 256));

  const float* xin = x0;
  for (int l = 0; l < 3; ++l) {
    const float* Wl  = (const float*)d_in[2 + 6 * l + 0];
    const float* bl  = (const float*)d_in[2 + 6 * l + 1];
    const float* Wr  = (const float*)d_in[2 + 6 * l + 2];
    const float* br  = (const float*)d_in[2 + 6 * l + 3];
    const float* att = (const float*)d_in[2 + 6 * l + 4];
    const float* bb  = (const float*)d_in[2 + 6 * l + 5];

    gemm128_wmma<<<gemmGrid, 256, 0, stream>>>(xin, Wl, bl, xl, nWaveTiles);
    gemm128_wmma<<<gemmGrid, 256, 0, stream>>>(xin, Wr, br, xr, nWaveTiles);
    init_layer<<<nodeGrid, 256, 0, stream>>>(m, z, acc);
    edge_scores<<<edgeGrid, 256, 0, stream>>>(ei, xl, xr, att, sw, m);
    edge_expsum<<<ehGrid, 256, 0, stream>>>(ei, sw, m, z);
    edge_aggregate<<<edgeGrid, 256, 0, stream>>>(ei, xl, sw, z, acc);

    float* out = (l == 2) ? (float*)d_out : xnx;
    finalize<<<nodeGrid, 256, 0, stream>>>(acc, bb, out, (l < 2) ? 1 : 0);
    xin = xnx;
  }
}